// FofeSeqEncoder_2671469658391
// MI455X (gfx1250) — compile-verified
//
#include <hip/hip_runtime.h>

// FOFE scan y_t = a*y_{t-1} + x_t, fp32, [B=8, T=4096, D=1024].
// Memory-bound: 268 MB min traffic (+58 MB halo) @ 23.3 TB/s -> ~14 us.
//  - 16 timesteps as Y = L*X + p*carry (L[m][k]=a^(m-k) lower-tri), via
//    4x V_WMMA_F32_16X16X4_F32 per (16t x 16d) sub-tile; carry via
//    ds_bpermute of D row 15 -> serial chain shortened 16x.
//  - Time split into 8 chunks of 512 + 256-step decay halo (a^256 ~ 2e-12).
//  - X staged global->LDS with GLOBAL_LOAD_ASYNC_TO_LDS_B128 (ASYNCcnt),
//    32-row stages, triple-buffered, 2 stages in flight; 256B-contiguous
//    row segments; B operands via ds_load_2addr_stride64.
//  - Stores: one base pointer per stage + immediate offsets (rows are
//    4096B-strided), avoiding per-row 64-bit address arithmetic.

#define ALPHA 0.9f
#define B_   8
#define T_   4096
#define D_   1024
#define NC_  8
#define L_   (T_ / NC_)   // 512
#define H_   256          // halo length (a^256 ~ 2e-12)
#define TR_  32           // rows per pipeline stage (2 WMMA sub-tiles)

typedef __attribute__((ext_vector_type(2))) float v2f;
typedef __attribute__((ext_vector_type(8))) float v8f;

__device__ __forceinline__ float alpha_pow(int e) {
  if (e < 0) return 0.0f;
  float v = 1.0f;
  for (int i = 0; i < e; ++i) v *= ALPHA;
  return v;
}

// One 16-row sub-tile: acc = L * X + cpow*carry; optionally store; new carry.
template <bool STORE>
__device__ __forceinline__ float subtile(const float* __restrict__ lpS,
                                         float* __restrict__ ytS, int hi, int n,
                                         v2f a0, v2f a1, v2f a2, v2f a3,
                                         const float cpow[8], float carry) {
  const float* lp = lpS + 2 * hi * 64;   // lane-half K rows {2,3} vs {0,1}
  v2f b0, b1, b2, b3;
  b0.x = lp[0 * 64];   b0.y = lp[1 * 64];
  b1.x = lp[4 * 64];   b1.y = lp[5 * 64];
  b2.x = lp[8 * 64];   b2.y = lp[9 * 64];
  b3.x = lp[12 * 64];  b3.y = lp[13 * 64];

  v8f acc;
#pragma unroll
  for (int r = 0; r < 8; ++r) acc[r] = cpow[r] * carry;

  acc = __builtin_amdgcn_wmma_f32_16x16x4_f32(false, a0, false, b0, (short)0, acc, false, false);
  acc = __builtin_amdgcn_wmma_f32_16x16x4_f32(false, a1, false, b1, (short)0, acc, false, false);
  acc = __builtin_amdgcn_wmma_f32_16x16x4_f32(false, a2, false, b2, (short)0, acc, false, false);
  acc = __builtin_amdgcn_wmma_f32_16x16x4_f32(false, a3, false, b3, (short)0, acc, false, false);

  if (STORE) {
#pragma unroll
    for (int r = 0; r < 8; ++r)
      ytS[r * D_] = acc[r];              // immediate offsets r*4096B
  }
  // New carry = D[15][n]; row 15 lives in VGPR 7 of lanes 16..31.
  return __shfl(acc[7], 16 + n, 32);
}

__global__ __launch_bounds__(128)
void fofe_wmma_kernel(const float* __restrict__ x, float* __restrict__ y) {
  __shared__ float lds[3][TR_ * 64];     // 3 x 8KB stage buffers

  const int tid  = threadIdx.x;
  const int lane = tid & 31;
  const int n    = lane & 15;
  const int hi   = lane >> 4;
  const int wib  = tid >> 5;

  // Block -> (batch, chunk, 64-dim group). 8*8*16 = 1024 blocks.
  const int dg  = blockIdx.x & 15;
  const int ck  = (blockIdx.x >> 4) & (NC_ - 1);
  const int bb  = blockIdx.x >> 7;
  const int d0g = dg << 6;
  const int d0l = wib << 4;

  // A operand (16x4 per WMMA): lane m=n holds A[m][k]; lane-half = k/2, reg = k%2.
  v2f a0, a1, a2, a3;
  {
    int k;
    k = 0  + 2 * hi;  a0.x = alpha_pow(n - k);  a0.y = alpha_pow(n - k - 1);
    k = 4  + 2 * hi;  a1.x = alpha_pow(n - k);  a1.y = alpha_pow(n - k - 1);
    k = 8  + 2 * hi;  a2.x = alpha_pow(n - k);  a2.y = alpha_pow(n - k - 1);
    k = 12 + 2 * hi;  a3.x = alpha_pow(n - k);  a3.y = alpha_pow(n - k - 1);
  }
  float cpow[8];
  cpow[0] = alpha_pow(8 * hi + 1);
#pragma unroll
  for (int r = 1; r < 8; ++r) cpow[r] = cpow[r - 1] * ALPHA;

  const size_t seqBase = (size_t)bb * T_ * D_;
  const float* gcol = x + seqBase + d0g;                         // stage-row base
  float*       yrow = y + seqBase + d0g + d0l + n + (size_t)(8 * hi) * D_;
  const uint32_t ldsBase = (uint32_t)(uintptr_t)&lds[0][0];
  const float*   ldsCol  = &lds[0][0] + d0l + n;

  // Per-thread async-copy segment constants: 128 threads x 4 x 16B = 8KB.
  int      segOff[4];    // float offset into stage (global side)
  uint32_t segLds[4];    // LDS byte address within buffer 0
#pragma unroll
  for (int j = 0; j < 4; ++j) {
    const int seg = tid + 128 * j;       // 0..511
    const int row = seg >> 4;            // 0..31
    const int c16 = seg & 15;
    segOff[j] = row * D_ + c16 * 4;
    segLds[j] = ldsBase + row * 256 + c16 * 16;
  }

  const int chunkStart = ck * L_;
  const int tFirst = (ck > 0) ? (chunkStart - H_) : 0;
  const int tLast  = chunkStart + L_ - TR_;        // last stage base

  // --- async stage issue -------------------------------------------------
  auto issue = [&](int tload, int buW) {
    const float* gt = gcol + (size_t)tload * D_;
#pragma unroll
    for (int j = 0; j < 4; ++j) {
      asm volatile("global_load_async_to_lds_b128 %0, %1, off"
                   :: "v"(segLds[j] + (uint32_t)(buW * (TR_ * 256))),
                      "v"((unsigned long long)(uintptr_t)(gt + segOff[j]))
                   : "memory");
    }
  };

  // Prime: stages 0 and 1 into buffers 0 and 1 (>= 16 stages always exist).
  issue(tFirst, 0);
  issue(tFirst + TR_, 1);

  float carry = 0.0f;
  int buR = 0, buW = 2;
  int tt = tFirst;

  // Halo stages: refine carry only.
  for (; tt < chunkStart; tt += TR_) {
    asm volatile("s_wait_asynccnt 0x4" ::: "memory");   // stage buR landed
    __syncthreads();
    int tload = tt + 2 * TR_;
    if (tload > tLast) tload = tLast;   // clamped dups hit dead buffers only
    issue(tload, buW);

    const float* lp = ldsCol + buR * (TR_ * 64);
    carry = subtile<false>(lp,            nullptr, hi, n, a0, a1, a2, a3, cpow, carry);
    carry = subtile<false>(lp + 16 * 64,  nullptr, hi, n, a0, a1, a2, a3, cpow, carry);
    buR = (buR == 2) ? 0 : buR + 1;
    buW = (buW == 2) ? 0 : buW + 1;
  }

  // Main stages: compute and store.
  for (; tt <= tLast; tt += TR_) {
    asm volatile("s_wait_asynccnt 0x4" ::: "memory");
    __syncthreads();
    int tload = tt + 2 * TR_;
    if (tload > tLast) tload = tLast;
    issue(tload, buW);

    const float* lp = ldsCol + buR * (TR_ * 64);
    float* yt = yrow + (size_t)tt * D_;
    carry = subtile<true>(lp,           yt,            hi, n, a0, a1, a2, a3, cpow, carry);
    carry = subtile<true>(lp + 16 * 64, yt + 16 * D_,  hi, n, a0, a1, a2, a3, cpow, carry);
    buR = (buR == 2) ? 0 : buR + 1;
    buW = (buW == 2) ? 0 : buW + 1;
  }
}

extern "C" void kernel_launch(void* const* d_in, const int* in_sizes, int n_in,
                              void* d_out, int out_size, void* d_ws, size_t ws_size,
                              hipStream_t stream) {
  (void)in_sizes; (void)n_in; (void)d_ws; (void)ws_size; (void)out_size;
  const float* x = (const float*)d_in[0];
  float*       y = (float*)d_out;

  dim3 grid(1024), block(128);
  hipLaunchKernelGGL(fofe_wmma_kernel, grid, block, 0, stream, x, y);
}